// SampleAndGroup_19396072308849
// MI455X (gfx1250) — compile-verified
//
#include <hip/hip_runtime.h>
#include <math.h>

// Problem constants (fixed by the reference)
#define B_   32
#define N_   32768
#define S_   2048
#define F_   64
#define TPB_FPS 1024
#define PPT  (N_ / TPB_FPS)   // 32 points per thread

typedef __attribute__((ext_vector_type(2))) float v2f;
typedef __attribute__((ext_vector_type(8))) float v8f;

// ---------------------------------------------------------------------------
// Kernel 1: Farthest-point sampling. One workgroup per batch (sequential in S).
// Coordinates and running min-distances live entirely in VGPRs; the inner loop
// is pure VALU. Argmax via wave32 shuffle reduce + tiny LDS cross-wave reduce.
// Writes sampled coords g[b][s][3] (sample s = "last" index before update,
// matching the reference scan: sample 0 is point 0).
// ---------------------------------------------------------------------------
__global__ __launch_bounds__(TPB_FPS) void fps_kernel(const float* __restrict__ x,
                                                      float* __restrict__ g)
{
    const int b   = blockIdx.x;
    const int tid = threadIdx.x;
    const float* xb = x + (size_t)b * 3u * N_;   // [3][N] channel-major
    const float* xx = xb;
    const float* xy = xb + N_;
    const float* xz = xb + 2 * N_;

    float px[PPT], py[PPT], pz[PPT], dist[PPT];
#pragma unroll
    for (int k = 0; k < PPT; ++k) {
        const int p = tid + k * TPB_FPS;
        px[k] = xx[p]; py[k] = xy[p]; pz[k] = xz[p];
        dist[k] = 1e10f;
    }

    __shared__ float s_v[TPB_FPS / 32];
    __shared__ int   s_i[TPB_FPS / 32];
    __shared__ float s_last[3];
    __shared__ int   s_bi;

    float* gb = g + (size_t)b * S_ * 3u;
    int last = 0;

    for (int s = 0; s < S_; ++s) {
        if (tid == 0) {
            const float lx = xx[last], ly = xy[last], lz = xz[last];
            s_last[0] = lx; s_last[1] = ly; s_last[2] = lz;
            gb[s * 3 + 0] = lx; gb[s * 3 + 1] = ly; gb[s * 3 + 2] = lz;
        }
        __syncthreads();
        const float lx = s_last[0], ly = s_last[1], lz = s_last[2];

        float bv = -1.0f; int bi = 0;
#pragma unroll
        for (int k = 0; k < PPT; ++k) {
            const float dx = px[k] - lx, dy = py[k] - ly, dz = pz[k] - lz;
            const float d  = dx * dx + dy * dy + dz * dz;
            const float nd = fminf(dist[k], d);
            dist[k] = nd;
            if (nd > bv) { bv = nd; bi = tid + k * TPB_FPS; }
        }
        // wave32 argmax reduce (tie -> smaller index, mimics argmax-first)
        for (int off = 16; off > 0; off >>= 1) {
            const float ov = __shfl_down(bv, off, 32);
            const int   oi = __shfl_down(bi, off, 32);
            if (ov > bv || (ov == bv && oi < bi)) { bv = ov; bi = oi; }
        }
        if ((tid & 31) == 0) { s_v[tid >> 5] = bv; s_i[tid >> 5] = bi; }
        __syncthreads();
        if (tid < 32) {
            bv = s_v[tid]; bi = s_i[tid];
            for (int off = 16; off > 0; off >>= 1) {
                const float ov = __shfl_down(bv, off, 32);
                const int   oi = __shfl_down(bi, off, 32);
                if (ov > bv || (ov == bv && oi < bi)) { bv = ov; bi = oi; }
            }
            if (tid == 0) s_bi = bi;
        }
        __syncthreads();
        last = s_bi;
    }
}

// ---------------------------------------------------------------------------
// Kernel 2: per-point linear 3->64 via V_WMMA_F32_16X16X4_F32.
// A (16x4): rows = 16 points, cols = (x,y,z,1)  -> bias folded into K=3.
// B (4x16): rows = (wx,wy,wz,bias) for 16 features of the N-tile.
// One wave per 16-point M-tile, looping the 4 feature tiles. 4096 waves total.
// ---------------------------------------------------------------------------
__global__ __launch_bounds__(256) void linear_wmma_kernel(const float* __restrict__ g,    // [B*S,3]
                                                          const float* __restrict__ w,    // [64,3]
                                                          const float* __restrict__ bias, // [64]
                                                          float* __restrict__ h)          // [B*S,64]
{
    const int lane  = threadIdx.x & 31;
    const int wave  = blockIdx.x * (blockDim.x >> 5) + (threadIdx.x >> 5);
    const int mbase = wave * 16;
    const int half  = lane >> 4;   // 0: lanes 0-15, 1: lanes 16-31
    const int l     = lane & 15;

    // A-matrix 16x4 f32: VGPR0/1 = K{0,1} (lanes 0-15), K{2,3} (lanes 16-31)
    const int row = mbase + l;
    v2f a;
    if (half == 0) { a.x = g[row * 3 + 0]; a.y = g[row * 3 + 1]; }
    else           { a.x = g[row * 3 + 2]; a.y = 1.0f; }

#pragma unroll
    for (int nt = 0; nt < 4; ++nt) {
        const int f = nt * 16 + l;
        v2f bm;
        if (half == 0) { bm.x = w[f * 3 + 0]; bm.y = w[f * 3 + 1]; }
        else           { bm.x = w[f * 3 + 2]; bm.y = bias[f]; }

        v8f c = {};
        // 8 args: (neg_a, A, neg_b, B, c_mod, C, reuse_a, reuse_b)
        c = __builtin_amdgcn_wmma_f32_16x16x4_f32(false, a, false, bm,
                                                  (short)0, c, false, false);

        const int col = nt * 16 + l;
#pragma unroll
        for (int v = 0; v < 8; ++v) {
            const int m = half ? (v + 8) : v;   // C/D layout: lanes16-31 carry M+8
            h[(size_t)(mbase + m) * F_ + col] = c[v];
        }
    }
}

// ---------------------------------------------------------------------------
// Kernel 3: BatchNorm batch statistics per feature (mean over B*S, biased var).
// ---------------------------------------------------------------------------
__global__ __launch_bounds__(256) void bn_stats_kernel(const float* __restrict__ h,
                                                       float* __restrict__ stats) // [2*F]
{
    const int f = blockIdx.x;
    const int tid = threadIdx.x;
    const int total = B_ * S_;
    float sum = 0.f, sq = 0.f;
    for (int i = tid; i < total; i += 256) {
        const float v = h[(size_t)i * F_ + f];
        sum += v; sq += v * v;
    }
    __shared__ float ss[256], s2[256];
    ss[tid] = sum; s2[tid] = sq;
    __syncthreads();
    for (int off = 128; off > 0; off >>= 1) {
        if (tid < off) { ss[tid] += ss[tid + off]; s2[tid] += s2[tid + off]; }
        __syncthreads();
    }
    if (tid == 0) {
        const float mu  = ss[0] / (float)total;
        const float var = s2[0] / (float)total - mu * mu;
        stats[f]      = mu;
        stats[F_ + f] = rsqrtf(var + 1e-5f);
    }
}

// ---------------------------------------------------------------------------
// Kernel 4: apply BN affine + LeakyReLU, then center-normalize over S
// (unbiased std, ddof=1) per (batch, feature), then alpha*h+beta. In place.
// One block per (b,f); each thread holds its 8 activations in registers
// between the reduction and the rewrite.
// ---------------------------------------------------------------------------
__global__ __launch_bounds__(256) void finalize_kernel(float* __restrict__ h,
                                                       const float* __restrict__ stats,
                                                       const float* __restrict__ gamma,
                                                       const float* __restrict__ bnb,
                                                       const float* __restrict__ alpha,
                                                       const float* __restrict__ beta)
{
    const int f = blockIdx.x & (F_ - 1);
    const int b = blockIdx.x >> 6;
    const int tid = threadIdx.x;

    const float mu = stats[f];
    const float rs = stats[F_ + f];
    const float ga = gamma[f], bb = bnb[f];

    const size_t base = ((size_t)b * S_) * F_ + f;
    float vals[S_ / 256];
    float sum = 0.f, sq = 0.f;
#pragma unroll
    for (int k = 0; k < S_ / 256; ++k) {
        const int s = k * 256 + tid;
        float v = h[base + (size_t)s * F_];
        v = ga * (v - mu) * rs + bb;         // BatchNorm
        v = (v >= 0.f) ? v : 0.01f * v;      // LeakyReLU
        vals[k] = v;
        sum += v; sq += v * v;
    }
    __shared__ float ss[256], s2[256];
    ss[tid] = sum; s2[tid] = sq;
    __syncthreads();
    for (int off = 128; off > 0; off >>= 1) {
        if (tid < off) { ss[tid] += ss[tid + off]; s2[tid] += s2[tid + off]; }
        __syncthreads();
    }
    const float m   = ss[0] / (float)S_;
    const float var = (s2[0] - (float)S_ * m * m) / (float)(S_ - 1); // ddof=1
    const float inv = 1.0f / (sqrtf(fmaxf(var, 0.f)) + 1e-5f);
    const float al = alpha[f], be = beta[f];
#pragma unroll
    for (int k = 0; k < S_ / 256; ++k) {
        const int s = k * 256 + tid;
        h[base + (size_t)s * F_] = (vals[k] - m) * inv * al + be;
    }
}

// ---------------------------------------------------------------------------
extern "C" void kernel_launch(void* const* d_in, const int* in_sizes, int n_in,
                              void* d_out, int out_size, void* d_ws, size_t ws_size,
                              hipStream_t stream)
{
    (void)in_sizes; (void)n_in; (void)out_size; (void)ws_size;
    const float* x        = (const float*)d_in[0];  // [B,3,N]
    const float* conv_w   = (const float*)d_in[1];  // [64,3]
    const float* conv_b   = (const float*)d_in[2];  // [64]
    const float* bn_gamma = (const float*)d_in[3];  // [64]
    const float* bn_beta  = (const float*)d_in[4];  // [64]
    const float* alpha    = (const float*)d_in[5];  // [1,1,64]
    const float* beta     = (const float*)d_in[6];  // [1,1,64]

    float* out   = (float*)d_out;                     // doubles as the h buffer [B,S,64]
    float* g     = (float*)d_ws;                      // [B,S,3] sampled points
    float* stats = g + (size_t)B_ * S_ * 3u;          // [2*64] BN mean / rsqrt(var+eps)

    fps_kernel<<<B_, TPB_FPS, 0, stream>>>(x, g);
    linear_wmma_kernel<<<(B_ * S_ / 16) / 8, 256, 0, stream>>>(g, conv_w, conv_b, out);
    bn_stats_kernel<<<F_, 256, 0, stream>>>(out, stats);
    finalize_kernel<<<B_ * F_, 256, 0, stream>>>(out, stats, bn_gamma, bn_beta, alpha, beta);
}